// GraphNeuralNetwork_81252191306417
// MI455X (gfx1250) — compile-verified
//
#include <hip/hip_runtime.h>
#include <hip/hip_bf16.h>

// ---------------------------------------------------------------------------
// GraphConv x2 on MI455X (gfx1250, wave32):
//   per layer: out = relu(x@Ws^T + bs + [cnt>0]*(nbr_mean@Wn^T + bn))
//   * scatter: 1 edge per wave32; float4/lane gather of x[src] (512B row),
//     hardware FP32 atomics into nbr_sum[dst] (node table lives in 192MB L2).
//   * fused layer GEMM: V_WMMA_F32_16X16X4_F32, 16x128 tile per wave.
//     B fragments batch-prefetched from LDS (separate Ws/Wn base pointers so
//     all ds loads use immediate offsets -> one wait per 16 WMMAs).
// ---------------------------------------------------------------------------

typedef __attribute__((ext_vector_type(2))) float v2f;
typedef __attribute__((ext_vector_type(8))) float v8f;

#define DIMF 128
#define LDW  130   // LDS row pitch in floats (even -> 8B-aligned v2f loads)

static __device__ __forceinline__ v8f wmma_f32_16x16x4(v2f a, v2f b, v8f c) {
    // D = A(16x4 f32) * B(4x16 f32) + C(16x16 f32)
    return __builtin_amdgcn_wmma_f32_16x16x4_f32(
        /*neg_a=*/false, a, /*neg_b=*/false, b,
        /*c_mod=*/(short)0, c, /*reuse_a=*/false, /*reuse_b=*/false);
}

// ---------------- zero nbr_sum + cnt (float4 grained) ------------------------
__global__ void gnn_zero_kernel(float4* __restrict__ p, long n4) {
    long i = (long)blockIdx.x * blockDim.x + threadIdx.x;
    if (i < n4) p[i] = make_float4(0.f, 0.f, 0.f, 0.f);
}

// ---------------- edge scatter: nbr_sum[dst] += x[src]; cnt[dst]++ -----------
__global__ void gnn_scatter_kernel(const float* __restrict__ xin,
                                   const int* __restrict__ dst,
                                   const int* __restrict__ src,
                                   float* __restrict__ nbr,
                                   float* __restrict__ cnt,
                                   int nEdges) {
    int wave = (int)((blockIdx.x * (long)blockDim.x + threadIdx.x) >> 5);
    int lane = threadIdx.x & 31;
    if (wave >= nEdges) return;
    int d = dst[wave];            // wave-uniform -> scalarized by compiler
    int s = src[wave];
    const float4* row = (const float4*)(xin + (long)s * DIMF);
    float4 v = row[lane];         // 32 lanes x 4 floats = 128-float row
    float* o = nbr + (long)d * DIMF + lane * 4;
    unsafeAtomicAdd(o + 0, v.x);  // -> global_atomic_add_f32 (L2 fp atomics)
    unsafeAtomicAdd(o + 1, v.y);
    unsafeAtomicAdd(o + 2, v.z);
    unsafeAtomicAdd(o + 3, v.w);
    if (lane == 0) unsafeAtomicAdd(cnt + d, 1.0f);
}

// ---------------- fused GraphConv layer (WMMA f32 16x16x4) -------------------
// Block: 256 threads = 8 waves; block covers 128 rows (16 rows per wave).
// Dynamic LDS: Ws then Wn, each 128 rows x LDW pitch floats.
__global__ void gnn_layer_kernel(const float* __restrict__ xin,
                                 const float* __restrict__ nbr,
                                 const float* __restrict__ cnt,
                                 const float* __restrict__ Ws,
                                 const float* __restrict__ bs,
                                 const float* __restrict__ Wn,
                                 const float* __restrict__ bn,
                                 float* __restrict__ out,
                                 int nNodes) {
    extern __shared__ float lds[];           // [2 * 128 * LDW]
    const int tid = threadIdx.x;

    // Stage both weight matrices (row-major [n][k]) into padded LDS.
    for (int i = tid; i < DIMF * DIMF; i += 256) {
        int n = i >> 7, k = i & 127;
        lds[n * LDW + k] = Ws[i];
        lds[DIMF * LDW + n * LDW + k] = Wn[i];
    }
    __syncthreads();

    const int wave = tid >> 5;
    const int lane = tid & 31;
    const int hi = lane >> 4;                // 0: K pair {0,1}, 1: K pair {2,3}
    const int lo = lane & 15;                // M (for A) / N (for B,C)

    const long base = (long)blockIdx.x * 128 + wave * 16;

    // A-fragment source row (clamped; tail rows are masked at store time).
    long arow = base + lo;
    if (arow >= nNodes) arow = 0;
    const float* axp = xin + arow * DIMF;
    const float* anp = nbr + arow * DIMF;
    float cdeg = cnt[arow];
    float rcnt = cdeg > 0.0f ? 1.0f / cdeg : 0.0f;   // mean + zero-neighbor mask

    // Per-wave LDS bases: all 8 n-tile fragments reachable via 16-bit DS
    // immediate offsets (t*16*LDW*4 = t*8320B <= 58240B) from each base.
    const float* wsbase0 = lds + lo * LDW;                 // Ws region
    const float* wnbase0 = lds + DIMF * LDW + lo * LDW;    // Wn region

    v8f acc[8] = {};                        // 8 n-tiles x (16x16 f32) = 16x128

    #pragma unroll 2
    for (int k0 = 0; k0 < DIMF; k0 += 4) {
        const int ka = k0 + 2 * hi;
        // A fragments (ISA 7.12.2: lane<16 holds K=k0..k0+1, lane>=16 K=k0+2..k0+3)
        v2f a_x = *(const v2f*)(axp + ka);
        v2f a_n = (*(const v2f*)(anp + ka)) * rcnt;

        // Batch-prefetch all 16 B fragments (one ds batch, one wait).
        v2f bsf[8], bnf[8];
        const float* ws = wsbase0 + ka;
        const float* wn = wnbase0 + ka;
        #pragma unroll
        for (int t = 0; t < 8; ++t) {
            bsf[t] = *(const v2f*)(ws + t * (16 * LDW));
            bnf[t] = *(const v2f*)(wn + t * (16 * LDW));
        }
        #pragma unroll
        for (int t = 0; t < 8; ++t) {
            acc[t] = wmma_f32_16x16x4(a_x, bsf[t], acc[t]);  // x @ Ws^T
            acc[t] = wmma_f32_16x16x4(a_n, bnf[t], acc[t]);  // nbr_mean @ Wn^T
        }
    }

    // Neighbor-bias mask per C-row: VGPR r holds row (base + r + 8*hi).
    float mask[8];
    #pragma unroll
    for (int r = 0; r < 8; ++r) {
        long rr = base + r + 8 * hi;
        mask[r] = (rr < nNodes && cnt[rr] > 0.0f) ? 1.0f : 0.0f;
    }

    #pragma unroll
    for (int t = 0; t < 8; ++t) {
        const int col = t * 16 + lo;
        const float bsv = bs[col];
        const float bnv = bn[col];
        #pragma unroll
        for (int r = 0; r < 8; ++r) {
            long rr = base + r + 8 * hi;
            if (rr < nNodes) {
                float v = acc[t][r] + bsv + mask[r] * bnv;
                out[rr * DIMF + col] = v > 0.0f ? v : 0.0f;
            }
        }
    }
}

// ---------------------------------------------------------------------------
extern "C" void kernel_launch(void* const* d_in, const int* in_sizes, int n_in,
                              void* d_out, int out_size, void* d_ws, size_t ws_size,
                              hipStream_t stream) {
    const float* x   = (const float*)d_in[0];
    const int*   ei  = (const int*)d_in[1];      // [2, E]: dst row then src row
    const float* Ws1 = (const float*)d_in[2];
    const float* bs1 = (const float*)d_in[3];
    const float* Wn1 = (const float*)d_in[4];
    const float* bn1 = (const float*)d_in[5];
    const float* Ws2 = (const float*)d_in[6];
    const float* bs2 = (const float*)d_in[7];
    const float* Wn2 = (const float*)d_in[8];
    const float* bn2 = (const float*)d_in[9];
    float* out = (float*)d_out;

    const int N = in_sizes[0] / DIMF;            // 40000
    const int E = in_sizes[1] / 2;               // 640000
    const int* dst = ei;
    const int* src = ei + E;

    // Workspace: h [N*D] | nbr_sum [N*D] | cnt [N] (N*D and N are mult. of 4)
    float* h    = (float*)d_ws;
    float* nbr  = h + (long)N * DIMF;
    float* cnt  = nbr + (long)N * DIMF;

    const long zero4 = ((long)N * DIMF + N) / 4; // nbr_sum + cnt contiguous
    const int  zeroB = (int)((zero4 + 255) / 256);
    const int  scatB = (E * 32 + 255) / 256;     // 1 wave (32 lanes) per edge
    const int  gemmB = (N + 127) / 128;          // 128 rows per block
    const size_t ldsBytes = 2u * DIMF * LDW * sizeof(float); // 133,120 B

    // -------- layer 1: x -> h --------
    gnn_zero_kernel<<<zeroB, 256, 0, stream>>>((float4*)nbr, zero4);
    gnn_scatter_kernel<<<scatB, 256, 0, stream>>>(x, dst, src, nbr, cnt, E);
    gnn_layer_kernel<<<gemmB, 256, ldsBytes, stream>>>(x, nbr, cnt,
                                                       Ws1, bs1, Wn1, bn1, h, N);
    // -------- layer 2: h -> out --------
    gnn_zero_kernel<<<zeroB, 256, 0, stream>>>((float4*)nbr, zero4);
    gnn_scatter_kernel<<<scatB, 256, 0, stream>>>(h, dst, src, nbr, cnt, E);
    gnn_layer_kernel<<<gemmB, 256, ldsBytes, stream>>>(h, nbr, cnt,
                                                       Ws2, bs2, Wn2, bn2, out, N);
}